// HybridSymmetricLoss_17600775979352
// MI455X (gfx1250) — compile-verified
//
#include <hip/hip_runtime.h>
#include <hip/hip_bf16.h>

typedef __attribute__((ext_vector_type(2))) float v2f;
typedef __attribute__((ext_vector_type(8))) float v8f;

namespace {
constexpr int kB   = 8192;   // batches
constexpr int kP   = 4;      // slots
constexpr int kMM  = 144;    // 12*12 elements per (b,p)
constexpr int kBatchesPerBlock = 16;   // 4 waves * 4 batches/wave
constexpr int kBlocks = kB / kBatchesPerBlock;  // 512
}

// itertools.permutations(range(4)) — lexicographic order (matters for argmin ties)
__device__ constexpr int PERMS[24][4] = {
  {0,1,2,3},{0,1,3,2},{0,2,1,3},{0,2,3,1},{0,3,1,2},{0,3,2,1},
  {1,0,2,3},{1,0,3,2},{1,2,0,3},{1,2,3,0},{1,3,0,2},{1,3,2,0},
  {2,0,1,3},{2,0,3,1},{2,1,0,3},{2,1,3,0},{2,3,0,1},{2,3,1,0},
  {3,0,1,2},{3,0,2,1},{3,1,0,2},{3,1,2,0},{3,2,0,1},{3,2,1,0}};

// ln(2) for converting v_log_f32 (log2) to natural log
#define LN2F 0.69314718055994531f

// clipped natural logs via raw hardware v_log_f32: trans + mul + max (3 VALU ops).
// Inputs are uniform in [0,1): no denormals; log(0) = -inf clips to -100 correctly.
__device__ __forceinline__ float clog(float x) {
    return fmaxf(__builtin_amdgcn_logf(x) * LN2F, -100.0f);
}
__device__ __forceinline__ float clog1m(float x) {
    return fmaxf(__builtin_amdgcn_logf(1.0f - x) * LN2F, -100.0f);
}

__global__ __launch_bounds__(128)
void hsl_main(const float* __restrict__ assign,
              const float* __restrict__ category,
              const float* __restrict__ alabels,
              const float* __restrict__ clabels,
              float* __restrict__ block_partials)
{
    __shared__ float sD[4][16][16];  // per-wave 16x16 WMMA result
    __shared__ float sS[4][16];      // per-wave sum(l1p) per row
    __shared__ float sWave[4];

    const int lane   = threadIdx.x & 31;
    const int wave   = threadIdx.x >> 5;
    const int b0     = blockIdx.x * kBatchesPerBlock + wave * 4;  // 4 batches per wave

    const int lane16 = lane & 15;
    const int half   = (lane >> 4) << 1;     // 0 (K=0,1) or 2 (K=2,3)
    const int bb     = lane16 >> 2;          // local batch 0..3
    const int slot   = lane16 & 3;           // i for A / j for B
    const int row    = (b0 + bb) * kP + slot;

    const float2* aptr = reinterpret_cast<const float2*>(assign  + (size_t)row * kMM + half);
    const float2* yptr = reinterpret_cast<const float2*>(alabels + (size_t)row * kMM + half);

    v8f acc = {};
    float sumL1p = 0.0f;

    // K loop: 144 / 4 = 36 WMMA accumulations of V_WMMA_F32_16X16X4_F32 (exact f32).
    // Unroll x4 so the compiler can batch global loads ahead of the WMMAs.
    #pragma unroll 4
    for (int t = 0; t < kMM / 4; ++t) {
        float2 a = aptr[t * 2];
        float2 y = yptr[t * 2];
        float lp0  = clog(a.x);
        float l1p0 = clog1m(a.x);
        float lp1  = clog(a.y);
        float l1p1 = clog1m(a.y);
        sumL1p += l1p0 + l1p1;
        v2f av; av.x = lp0 - l1p0; av.y = lp1 - l1p1;   // d = lp - l1p
        v2f bv; bv.x = y.x;        bv.y = y.y;          // labels (0/1)
        // D += A(16x4 f32) x B(4x16 f32); 8-arg form: neg_a, A, neg_b, B, c_mod, C, reuse_a, reuse_b
        acc = __builtin_amdgcn_wmma_f32_16x16x4_f32(
            false, av, false, bv, (short)0, acc, false, false);
    }

    // combine the two K-half partials of sum(l1p): lanes L and L^16 share a row
    sumL1p += __shfl_xor(sumL1p, 16, 32);

    // spill D (C/D layout: VGPR k -> M = k + 8*(lane>=16), N = lane&15) and S to LDS
    const int mbase = (lane >> 4) * 8;
    #pragma unroll
    for (int k = 0; k < 8; ++k)
        sD[wave][mbase + k][lane16] = acc[k];
    if (lane < 16) sS[wave][lane16] = sumL1p;
    __syncthreads();

    // finalize: lanes 0..3 each own one local batch
    float partial = 0.0f;
    if (lane < 4) {
        const int lb  = lane;                 // local batch
        const int myb = b0 + lb;
        float cost[4][4];
        #pragma unroll
        for (int i = 0; i < 4; ++i) {
            float S = sS[wave][4 * lb + i];
            #pragma unroll
            for (int j = 0; j < 4; ++j)
                cost[i][j] = -(sD[wave][4 * lb + i][4 * lb + j] + S) * (1.0f / (float)kMM);
        }
        float best = 1e30f;
        int bestp = 0;   // packed winning permutation
        #pragma unroll
        for (int k = 0; k < 24; ++k) {
            float l = cost[PERMS[k][0]][0] + cost[PERMS[k][1]][1] +
                      cost[PERMS[k][2]][2] + cost[PERMS[k][3]][3];
            int packed = PERMS[k][0] | (PERMS[k][1] << 2) |
                         (PERMS[k][2] << 4) | (PERMS[k][3] << 6);
            if (l < best) { best = l; bestp = packed; }   // strict <: first-index argmin
        }
        best *= 0.25f;

        float catl = 0.0f;
        #pragma unroll
        for (int p = 0; p < 4; ++p) {
            int src  = (bestp >> (2 * p)) & 3;
            float c  = category[myb * kP + src];
            float yl = clabels[myb * kP + p];
            catl += -(yl * clog(c) + (1.0f - yl) * clog1m(c));
        }
        // W_ASSIGN * mean_b(min) + W_CAT * mean_{b,p}(catBCE)
        partial = best * (1.0f / (float)kB) + catl * (0.5f / (float)(kB * kP));
    }
    // sum lanes 0..3 into lane 0 (fixed pairing -> deterministic)
    partial += __shfl_xor(partial, 1, 32);
    partial += __shfl_xor(partial, 2, 32);
    if (lane == 0) sWave[wave] = partial;
    __syncthreads();
    if (threadIdx.x == 0)
        block_partials[blockIdx.x] = (sWave[0] + sWave[1]) + (sWave[2] + sWave[3]);
}

__global__ __launch_bounds__(256)
void hsl_reduce(const float* __restrict__ partials, float* __restrict__ out, int n)
{
    __shared__ float s[256];
    int t = threadIdx.x;
    float v = 0.0f;
    for (int i = t; i < n; i += 256) v += partials[i];   // fixed order per thread
    s[t] = v;
    __syncthreads();
    #pragma unroll
    for (int stride = 128; stride > 0; stride >>= 1) {
        if (t < stride) s[t] += s[t + stride];
        __syncthreads();
    }
    if (t == 0) out[0] = s[0];
}

extern "C" void kernel_launch(void* const* d_in, const int* in_sizes, int n_in,
                              void* d_out, int out_size, void* d_ws, size_t ws_size,
                              hipStream_t stream) {
    const float* assign   = (const float*)d_in[0];  // (8192,4,12,12)
    const float* category = (const float*)d_in[1];  // (8192,4)
    const float* alabels  = (const float*)d_in[2];  // (8192,4,12,12)
    const float* clabels  = (const float*)d_in[3];  // (8192,4)
    float* partials = (float*)d_ws;                 // 512 floats

    hsl_main<<<kBlocks, 128, 0, stream>>>(assign, category, alabels, clabels, partials);
    hsl_reduce<<<1, 256, 0, stream>>>(partials, (float*)d_out, kBlocks);
}